// EinsumTensorProduct_11536282157110
// MI455X (gfx1250) — compile-verified
//
#include <hip/hip_runtime.h>

typedef __attribute__((ext_vector_type(2))) float v2f;
typedef __attribute__((ext_vector_type(8))) float v8f;

#define DIM   9     // features per input vector
#define NK    81    // output features (k3)
#define KIJ   81    // contraction length (i*9+j)
#define KPAD  84    // K padded to multiple of 4
#define NPAD  96    // N padded to 6 tiles of 16
#define WSTR  84    // LDS row stride in floats (even -> 8B-aligned rows)

__global__ __launch_bounds__(256)
void etp_wmma_f32_kernel(const float* __restrict__ in1,
                         const float* __restrict__ in2,
                         const float* __restrict__ cb,
                         float* __restrict__ out,
                         int nb)
{
    // Wt[col][k] = cb[col*81 + k]  (cb flat [k3][kij] is exactly the B-matrix
    // transposed-for-column-access layout we want), zero padded to 96x84.
    __shared__ float Wt[NPAD * WSTR];   // 31.5 KB

    for (int idx = threadIdx.x; idx < NPAD * WSTR; idx += blockDim.x) {
        int col = idx / WSTR;
        int k   = idx - col * WSTR;
        float v = 0.0f;
        if (col < NK && k < KIJ) v = cb[col * KIJ + k];
        Wt[idx] = v;
    }
    __syncthreads();

    const int lane  = threadIdx.x & 31;
    const int wave  = threadIdx.x >> 5;
    const int wpb   = blockDim.x >> 5;
    const int gwave = blockIdx.x * wpb + wave;
    const int nwav  = gridDim.x * wpb;
    const int row   = lane & 15;          // A-matrix row / C-matrix column idx
    const bool lo   = lane < 16;          // K-phase: lanes<16 -> K%4 in {0,1}

    const int ntiles = nb >> 4;           // 16 batch rows per tile

    for (int tile = gwave; tile < ntiles; tile += nwav) {
        const int b = (tile << 4) + row;  // this lane's batch row
        const float* p1 = in1 + (size_t)b * DIM;
        const float* p2 = in2 + (size_t)b * DIM;
        float av[DIM], bv[DIM];
#pragma unroll
        for (int q = 0; q < DIM; ++q) { av[q] = p1[q]; bv[q] = p2[q]; }

        // Build all 21 A fragments (16x4 f32 each) in registers.
        // lane<16 holds K = {4s, 4s+1}; lane>=16 holds K = {4s+2, 4s+3}.
        v2f af[KPAD / 4];
#pragma unroll
        for (int s = 0; s < KPAD / 4; ++s) {
            float e[4];
#pragma unroll
            for (int u = 0; u < 4; ++u) {
                int k = 4 * s + u;                      // compile-time constant
                e[u] = (k < KIJ) ? av[k / DIM] * bv[k % DIM] : 0.0f;
            }
            af[s].x = lo ? e[0] : e[2];
            af[s].y = lo ? e[1] : e[3];
        }

        // 6 output tiles of 16 columns each: out_tile = P(16x84) x W(84x16)
#pragma unroll
        for (int n = 0; n < NPAD / 16; ++n) {
            v8f acc = {};
            const int col = n * 16 + row;               // B-matrix column
            const float* wrow = &Wt[col * WSTR + (lo ? 0 : 2)];
#pragma unroll
            for (int s = 0; s < KPAD / 4; ++s) {
                v2f bf;
                bf.x = wrow[4 * s];                     // K = 4s + c0
                bf.y = wrow[4 * s + 1];                 // K = 4s + c0 + 1
                acc = __builtin_amdgcn_wmma_f32_16x16x4_f32(
                    false, af[s], false, bf,
                    (short)0, acc, false, false);
            }
            if (col < NK) {
                const size_t base = (size_t)(tile << 4) * NK + col;
#pragma unroll
                for (int v = 0; v < 8; ++v) {
                    int r = v + (lo ? 0 : 8);           // C layout: M = v (+8 hi)
                    out[base + (size_t)r * NK] = acc[v];
                }
            }
        }
    }
}

extern "C" void kernel_launch(void* const* d_in, const int* in_sizes, int n_in,
                              void* d_out, int out_size, void* d_ws, size_t ws_size,
                              hipStream_t stream) {
    const float* in1 = (const float*)d_in[0];
    const float* in2 = (const float*)d_in[1];
    const float* cb  = (const float*)d_in[2];
    float* out = (float*)d_out;
    const int nb = in_sizes[0] / DIM;     // batch size (1048576)

    dim3 grid(2048), block(256);          // 16384 waves, 4 batch tiles each
    etp_wmma_f32_kernel<<<grid, block, 0, stream>>>(in1, in2, cb, out, nb);
}